// GraphNN_14594298872170
// MI455X (gfx1250) — compile-verified
//
#include <hip/hip_runtime.h>

#define D 128

typedef __attribute__((ext_vector_type(2))) float v2f;
typedef __attribute__((ext_vector_type(8))) float v8f;

__device__ __forceinline__ void atomAddF(float* p, float v) {
    // hardware global_atomic_add_f32 (no CAS loop)
    unsafeAtomicAdd(p, v);
}

// ---- degree / normalization -------------------------------------------------
__global__ void k_deg_init(float* __restrict__ deg, int n) {
    int i = blockIdx.x * blockDim.x + threadIdx.x;
    if (i < n) deg[i] = 1.0f;                    // self-loop
}

__global__ void k_deg_acc(const int* __restrict__ dst, float* __restrict__ deg, int e) {
    int i = blockIdx.x * blockDim.x + threadIdx.x;
    if (i < e) atomAddF(&deg[dst[i]], 1.0f);
}

__global__ void k_rsqrt(float* __restrict__ deg, int n) {
    int i = blockIdx.x * blockDim.x + threadIdx.x;
    if (i < n) deg[i] = 1.0f / sqrtf(deg[i]);    // deg >= 1 always (self-loops)
}

// ---- conv1: scatter  agg[dst] += dis[s]*dis[t]*ea[s] * W1[:]  ---------------
// one wave (32 lanes) per edge; each lane owns 4 consecutive features
__global__ void k_scatter1(const int* __restrict__ src, const int* __restrict__ dst,
                           const float* __restrict__ dis, const float* __restrict__ ea,
                           const float* __restrict__ W1, float* __restrict__ agg, int e) {
    int edge = blockIdx.x * (blockDim.x >> 5) + (threadIdx.x >> 5);
    if (edge >= e) return;
    int lane = threadIdx.x & 31;
    int s = src[edge], t = dst[edge];
    float w = dis[s] * dis[t] * ea[s];
    int d = lane * 4;
    float4 w1 = *(const float4*)(W1 + d);
    float* p = agg + (size_t)t * D + d;
    atomAddF(p + 0, w * w1.x);
    atomAddF(p + 1, w * w1.y);
    atomAddF(p + 2, w * w1.z);
    atomAddF(p + 3, w * w1.w);
}

// h1 = relu(agg + self_loop + b1)  (in place)
__global__ void k_bias_relu1(float* __restrict__ agg, const float* __restrict__ ea,
                             const float* __restrict__ dis, const float* __restrict__ W1,
                             const float* __restrict__ b1, int n) {
    int idx = blockIdx.x * blockDim.x + threadIdx.x;
    if (idx >= n * D) return;
    int i = idx >> 7, d = idx & (D - 1);
    float di = dis[i];
    float v = agg[idx] + ea[i] * di * di * W1[d] + b1[d];
    agg[idx] = fmaxf(v, 0.0f);
}

// ---- conv2 GEMM:  Bout[Npad x 128] = A[Npad x 128] @ W[128 x 128]  ----------
// fp32 WMMA 16x16x4; one wave per 16x16 output tile, 8 waves/block -> 16x128
__global__ void k_gemm(const float* __restrict__ A, const float* __restrict__ W,
                       float* __restrict__ Bout) {
    const int lane = threadIdx.x & 31;
    const int wv   = threadIdx.x >> 5;       // 0..7 -> output col tile
    const int m0   = blockIdx.x * 16;
    const int n0   = wv * 16;
    const int mn   = lane & 15;              // m for A, n for B/C
    const int h    = lane >> 4;              // K-half selector
    const float* arow = A + (size_t)(m0 + mn) * D;
    v8f c = {};
#pragma unroll
    for (int kb = 0; kb < D; kb += 4) {
        // A 16x4: VGPR0: K=2h, VGPR1: K=2h+1 (per ISA layout)
        v2f a = *(const v2f*)(arow + kb + h * 2);
        // B 4x16: mirrored layout, column n per lane
        v2f b;
        b.x = W[(size_t)(kb + h * 2 + 0) * D + n0 + mn];
        b.y = W[(size_t)(kb + h * 2 + 1) * D + n0 + mn];
        c = __builtin_amdgcn_wmma_f32_16x16x4_f32(
                /*neg_a=*/false, a, /*neg_b=*/false, b,
                /*c_mod=*/(short)0, c, /*reuse_a=*/false, /*reuse_b=*/false);
    }
    // C/D layout: VGPR r -> row m0 + 8*h + r, col n0 + (lane&15)
    const int mbase = m0 + h * 8;
#pragma unroll
    for (int r = 0; r < 8; ++r)
        Bout[(size_t)(mbase + r) * D + n0 + mn] = c[r];
}

// ---- conv2 scatter:  agg[dst] += dis[s]*dis[t] * t[s][:]  -------------------
__global__ void k_scatter2(const int* __restrict__ src, const int* __restrict__ dst,
                           const float* __restrict__ dis, const float* __restrict__ t,
                           float* __restrict__ agg, int e) {
    int edge = blockIdx.x * (blockDim.x >> 5) + (threadIdx.x >> 5);
    if (edge >= e) return;
    int lane = threadIdx.x & 31;
    int s = src[edge], dd = dst[edge];
    float w = dis[s] * dis[dd];
    int d = lane * 4;
    float4 tv = *(const float4*)(t + (size_t)s * D + d);
    float* p = agg + (size_t)dd * D + d;
    atomAddF(p + 0, w * tv.x);
    atomAddF(p + 1, w * tv.y);
    atomAddF(p + 2, w * tv.z);
    atomAddF(p + 3, w * tv.w);
}

// h2 = relu(agg + t*dis^2 + b2), fused with mean-pool column sums
#define POOL_ROWS 128
__global__ void k_fuse_pool(const float* __restrict__ agg, const float* __restrict__ t,
                            const float* __restrict__ dis, const float* __restrict__ b2,
                            float* __restrict__ pooled, int n) {
    int d  = threadIdx.x;                    // 0..127
    int r0 = blockIdx.x * POOL_ROWS;
    int r1 = min(r0 + POOL_ROWS, n);
    float bb = b2[d];
    float local = 0.0f;
    for (int r = r0; r < r1; ++r) {
        float di = dis[r];
        float v  = agg[(size_t)r * D + d] + t[(size_t)r * D + d] * di * di + bb;
        local += fmaxf(v, 0.0f);
    }
    atomAddF(&pooled[d], local);
}

// pooled/N -> fc1+relu -> fc2 -> out[2]
__global__ void k_head(const float* __restrict__ pooled,
                       const float* __restrict__ fc1w, const float* __restrict__ fc1b,
                       const float* __restrict__ fc2w, const float* __restrict__ fc2b,
                       float* __restrict__ out, int n) {
    __shared__ float sp[D];
    __shared__ float sz[D];
    int d = threadIdx.x;
    sp[d] = pooled[d] / (float)n;
    __syncthreads();
    float acc = fc1b[d];
    for (int k = 0; k < D; ++k) acc += sp[k] * fc1w[k * D + d];
    sz[d] = fmaxf(acc, 0.0f);
    __syncthreads();
    if (d < 2) {
        float o = fc2b[d];
        for (int k = 0; k < D; ++k) o += sz[k] * fc2w[k * 2 + d];
        out[d] = o;
    }
}

extern "C" void kernel_launch(void* const* d_in, const int* in_sizes, int n_in,
                              void* d_out, int out_size, void* d_ws, size_t ws_size,
                              hipStream_t stream) {
    const int*   edge_index = (const int*)d_in[1];
    const float* ea   = (const float*)d_in[2];
    const float* W1   = (const float*)d_in[3];
    const float* b1   = (const float*)d_in[4];
    const float* W2   = (const float*)d_in[5];
    const float* b2   = (const float*)d_in[6];
    const float* fc1w = (const float*)d_in[7];
    const float* fc1b = (const float*)d_in[8];
    const float* fc2w = (const float*)d_in[9];
    const float* fc2b = (const float*)d_in[10];

    const int N = in_sizes[2];          // edge_attr rows = node count
    const int E = in_sizes[1] / 2;      // edge_index is (2, E)
    const int* src = edge_index;
    const int* dst = edge_index + E;

    const int    Npad = (N + 15) & ~15;
    const size_t nd   = (size_t)Npad * D;

    float* dis    = (float*)d_ws;                       // Npad (rounded) floats
    float* bufA   = dis  + (((size_t)Npad + 63) & ~63ull);
    float* bufB   = bufA + nd;
    float* pooled = bufB + nd;                          // D floats

    // normalization: deg (with self-loops) -> dis = deg^-1/2
    k_deg_init<<<(N + 255) / 256, 256, 0, stream>>>(dis, N);
    k_deg_acc <<<(E + 255) / 256, 256, 0, stream>>>(dst, dis, E);
    k_rsqrt   <<<(N + 255) / 256, 256, 0, stream>>>(dis, N);

    // conv1 (linear fused into scatter since in_features == 1)
    hipMemsetAsync(bufA, 0, nd * sizeof(float), stream);
    k_scatter1<<<(E + 7) / 8, 256, 0, stream>>>(src, dst, dis, ea, W1, bufA, E);
    k_bias_relu1<<<((size_t)N * D + 255) / 256, 256, 0, stream>>>(bufA, ea, dis, W1, b1, N);

    // conv2: t = h1 @ W2 (WMMA fp32), then scatter + fused bias/relu/pool
    k_gemm<<<Npad / 16, 256, 0, stream>>>(bufA, W2, bufB);
    hipMemsetAsync(bufA, 0, nd * sizeof(float), stream);
    hipMemsetAsync(pooled, 0, D * sizeof(float), stream);
    k_scatter2<<<(E + 7) / 8, 256, 0, stream>>>(src, dst, dis, bufB, bufA, E);
    k_fuse_pool<<<(N + POOL_ROWS - 1) / POOL_ROWS, D, 0, stream>>>(bufA, bufB, dis, b2, pooled, N);

    // head
    k_head<<<1, D, 0, stream>>>(pooled, fc1w, fc1b, fc2w, fc2b, (float*)d_out, N);
}